// GCN_14431090114865
// MI455X (gfx1250) — compile-verified
//
#include <hip/hip_runtime.h>

typedef __attribute__((ext_vector_type(2))) float v2f;
typedef __attribute__((ext_vector_type(8))) float v8f;

#define IN_DIM 128
#define HID    64
#define NCLS   40

__device__ __forceinline__ void atomic_add_f32(float* p, float v) {
  // HW global_atomic_add_f32 (no CAS loop)
  unsafeAtomicAdd(p, v);
}

// ---------------- degree / normalization ----------------
__global__ void k_init_deg(float* __restrict__ deg, int n) {
  int i = blockIdx.x * blockDim.x + threadIdx.x;
  if (i < n) deg[i] = 1.0f;  // self-loop contribution
}

__global__ void k_deg(const long long* __restrict__ dst, float* __restrict__ deg, int e) {
  int i = blockIdx.x * blockDim.x + threadIdx.x;
  if (i < e) atomic_add_f32(&deg[(int)dst[i]], 1.0f);
}

__global__ void k_dinv(float* __restrict__ deg, int n) {
  int i = blockIdx.x * blockDim.x + threadIdx.x;
  if (i < n) deg[i] = rsqrtf(deg[i]);  // deg >= 1 always
}

// ---------------- GEMM1: h1 = x @ W1  (WMMA f32 16x16x4) ----------------
// one wave per 16x16 output tile; grid.x = (N/16)*4
__global__ void __launch_bounds__(32) k_gemm1(const float* __restrict__ x,
                                              const float* __restrict__ w,
                                              float* __restrict__ h1) {
  const int lane = threadIdx.x;
  const int lo = lane & 15, hi = lane >> 4;
  const int mt = blockIdx.x >> 2;
  const int nt = blockIdx.x & 3;
  const int m0 = mt << 4, n0 = nt << 4;

  // A: lane holds row (m0+lo), K elements {k+2*hi, k+2*hi+1}
  const float* arow = x + (size_t)(m0 + lo) * IN_DIM + 2 * hi;
  // B: v0 = W[(k+2*hi)][n0+lo], v1 = W[(k+2*hi+1)][n0+lo]
  const float* bcol = w + (size_t)(2 * hi) * HID + n0 + lo;

  v8f c = {};
#pragma unroll
  for (int k = 0; k < IN_DIM; k += 4) {
    v2f a = *(const v2f*)(arow + k);
    v2f b;
    b.x = bcol[k * HID];
    b.y = bcol[k * HID + HID];
    c = __builtin_amdgcn_wmma_f32_16x16x4_f32(false, a, false, b, (short)0, c, false, false);
  }
  // D: VGPR i -> row m0 + i + 8*hi, col n0+lo
  float* o = h1 + (size_t)(m0 + 8 * hi) * HID + n0 + lo;
#pragma unroll
  for (int i = 0; i < 8; ++i) o[(size_t)i * HID] = c[i];
}

// ---------------- GEMM2: h2 = relu(agg1) @ W2 (N=40 padded to 48) ----------------
// grid.x = (N/16)*3
__global__ void __launch_bounds__(32) k_gemm2(const float* __restrict__ g,
                                              const float* __restrict__ w,
                                              float* __restrict__ h2) {
  const int lane = threadIdx.x;
  const int lo = lane & 15, hi = lane >> 4;
  const int mt = blockIdx.x / 3;
  const int nt = blockIdx.x % 3;
  const int m0 = mt << 4;
  const int col = nt * 16 + lo;
  const int cc = col < NCLS ? col : 0;  // clamp loads in-bounds; junk cols never stored

  const float* arow = g + (size_t)(m0 + lo) * HID + 2 * hi;
  const float* bcol = w + (size_t)(2 * hi) * NCLS + cc;

  v8f c = {};
#pragma unroll
  for (int k = 0; k < HID; k += 4) {
    v2f a = *(const v2f*)(arow + k);
    a.x = fmaxf(a.x, 0.0f);  // fused ReLU
    a.y = fmaxf(a.y, 0.0f);
    v2f b;
    b.x = bcol[k * NCLS];
    b.y = bcol[k * NCLS + NCLS];
    c = __builtin_amdgcn_wmma_f32_16x16x4_f32(false, a, false, b, (short)0, c, false, false);
  }
  if (col < NCLS) {
    float* o = h2 + (size_t)(m0 + 8 * hi) * NCLS + col;
#pragma unroll
    for (int i = 0; i < 8; ++i) o[(size_t)i * NCLS] = c[i];
  }
}

// ---------------- self-loop init + per-edge scatter (layer 1, F=64) ----------------
__global__ void k_selfinit1(const float* __restrict__ h1, const float* __restrict__ dinv,
                            const float* __restrict__ b1, float* __restrict__ agg, int n) {
  int t = blockIdx.x * blockDim.x + threadIdx.x;
  if (t >= n * HID) return;
  int i = t >> 6, f = t & 63;
  float di = dinv[i];
  agg[t] = b1[f] + di * di * h1[t];
}

__global__ void k_edge1(const long long* __restrict__ src, const long long* __restrict__ dst,
                        const float* __restrict__ h1, const float* __restrict__ dinv,
                        float* __restrict__ agg, int e) {
  int t = blockIdx.x * blockDim.x + threadIdx.x;
  if (t >= e * HID) return;  // 102.4M < 2^31
  int ed = t >> 6, f = t & 63;
  int s = (int)src[ed], d = (int)dst[ed];
  float cf = dinv[s] * dinv[d];
  atomic_add_f32(&agg[(size_t)d * HID + f], cf * h1[(size_t)s * HID + f]);
}

// ---------------- self-loop init + per-edge scatter (layer 2, F=40) ----------------
__global__ void k_selfinit2(const float* __restrict__ h2, const float* __restrict__ dinv,
                            const float* __restrict__ b2, float* __restrict__ out, int n) {
  int t = blockIdx.x * blockDim.x + threadIdx.x;
  if (t >= n * NCLS) return;
  unsigned ut = (unsigned)t;
  int i = ut / NCLS, f = ut % NCLS;
  float di = dinv[i];
  out[t] = b2[f] + di * di * h2[t];
}

__global__ void k_edge2(const long long* __restrict__ src, const long long* __restrict__ dst,
                        const float* __restrict__ h2, const float* __restrict__ dinv,
                        float* __restrict__ out, int e) {
  int t = blockIdx.x * blockDim.x + threadIdx.x;
  if (t >= e * NCLS) return;  // 64M
  unsigned ut = (unsigned)t;
  int ed = ut / NCLS, f = ut % NCLS;
  int s = (int)src[ed], d = (int)dst[ed];
  float cf = dinv[s] * dinv[d];
  atomic_add_f32(&out[(size_t)d * NCLS + f], cf * h2[(size_t)s * NCLS + f]);
}

extern "C" void kernel_launch(void* const* d_in, const int* in_sizes, int n_in,
                              void* d_out, int out_size, void* d_ws, size_t ws_size,
                              hipStream_t stream) {
  const float*     x  = (const float*)d_in[0];
  const long long* ei = (const long long*)d_in[1];
  const float*     W1 = (const float*)d_in[2];
  const float*     b1 = (const float*)d_in[3];
  const float*     W2 = (const float*)d_in[4];
  const float*     b2 = (const float*)d_in[5];
  float* out = (float*)d_out;

  const int N = in_sizes[0] / IN_DIM;  // 100000 (multiple of 16)
  const int E = in_sizes[1] / 2;       // 1600000
  const long long* src = ei;
  const long long* dst = ei + E;

  // workspace: dinv[N] | h1[N*64] (aliased later by h2[N*40]) | agg1[N*64]
  float* ws   = (float*)d_ws;
  float* dinv = ws;
  size_t off  = ((size_t)N + 255) & ~(size_t)255;
  float* h1   = ws + off;
  float* agg  = h1 + (size_t)N * HID;
  float* h2   = h1;  // h1 dead after k_edge1

  const int B = 256;
  k_init_deg<<<(N + B - 1) / B, B, 0, stream>>>(dinv, N);
  k_deg<<<(E + B - 1) / B, B, 0, stream>>>(dst, dinv, E);
  k_dinv<<<(N + B - 1) / B, B, 0, stream>>>(dinv, N);

  k_gemm1<<<(N / 16) * 4, 32, 0, stream>>>(x, W1, h1);
  k_selfinit1<<<(N * HID + B - 1) / B, B, 0, stream>>>(h1, dinv, b1, agg, N);
  k_edge1<<<(E * HID + B - 1) / B, B, 0, stream>>>(src, dst, h1, dinv, agg, E);

  k_gemm2<<<(N / 16) * 3, 32, 0, stream>>>(agg, W2, h2);
  k_selfinit2<<<(N * NCLS + B - 1) / B, B, 0, stream>>>(h2, dinv, b2, out, N);
  k_edge2<<<(E * NCLS + B - 1) / B, B, 0, stream>>>(src, dst, h2, dinv, out, E);
}